// Enblock_2267742732490
// MI455X (gfx1250) — compile-verified
//
#include <hip/hip_runtime.h>
#include <math.h>

typedef __attribute__((ext_vector_type(2))) float v2f;
typedef __attribute__((ext_vector_type(8))) float v8f;

#define BATCH 16
#define CIN   32
#define COUT  64
#define KCHEB 6

// ---------------- degree / norm ----------------
__global__ void deg_count(const int* __restrict__ erow, float* __restrict__ deg, int nE) {
  int t = blockIdx.x * blockDim.x + threadIdx.x;
  if (t < nE) atomicAdd(&deg[erow[t]], 1.0f);
}

__global__ void deg_to_dinv(float* __restrict__ deg, int nN) {
  int t = blockIdx.x * blockDim.x + threadIdx.x;
  if (t < nN) { float d = deg[t]; deg[t] = d > 0.f ? rsqrtf(d) : 0.f; }
}

__global__ void edge_norm(const int* __restrict__ erow, const int* __restrict__ ecol,
                          const float* __restrict__ dinv, float* __restrict__ norm, int nE) {
  int t = blockIdx.x * blockDim.x + threadIdx.x;
  if (t < nE) norm[t] = -dinv[erow[t]] * dinv[ecol[t]];
}

// ---------------- SpMM scatter: dst[b,row,:] += scale*norm[e]*src[b,col,:] ----------------
// 8 consecutive threads cover one (edge,batch) row of 32 channels (float4 each):
// coalesced 128B gathers and grouped atomics.
__global__ void spmm_scatter(const float* __restrict__ src, float* __restrict__ dst,
                             const int* __restrict__ erow, const int* __restrict__ ecol,
                             const float* __restrict__ norm, float scale, int nE, int nN) {
  long long t = (long long)blockIdx.x * blockDim.x + threadIdx.x;
  long long total = (long long)nE * BATCH * 8;
  if (t >= total) return;
  int c = (int)(t & 7) * 4;
  long long eb = t >> 3;
  int e = (int)(eb % nE);
  int b = (int)(eb / nE);
  float w = norm[e] * scale;
  const float4 v = *(const float4*)(src + ((size_t)b * nN + ecol[e]) * CIN + c);
  float* d = dst + ((size_t)b * nN + erow[e]) * CIN + c;
  atomicAdd(d + 0, v.x * w);
  atomicAdd(d + 1, v.y * w);
  atomicAdd(d + 2, v.z * w);
  atomicAdd(d + 3, v.w * w);
}

// nxt = nxt - prv (elementwise, float4)
__global__ void axpy_sub(float* __restrict__ nxt, const float* __restrict__ prv, long long n4) {
  long long t = (long long)blockIdx.x * blockDim.x + threadIdx.x;
  if (t >= n4) return;
  float4 a = ((const float4*)nxt)[t];
  float4 b = ((const float4*)prv)[t];
  a.x -= b.x; a.y -= b.y; a.z -= b.z; a.w -= b.w;
  ((float4*)nxt)[t] = a;
}

// ---------------- WMMA GEMM: H[rows,64] (+)= A[rows,32] @ Wk[32,64] ----------------
// One wave -> 16 rows x 64 cols. K loop = 8 steps of V_WMMA_F32_16X16X4_F32 per 16-col tile.
// W is staged into LDS K-pair interleaved: sW[(kr/2)*COUT + col][2] = {W[kr][col], W[kr+1][col]}
// so each B fragment is a single aligned 8-byte ds load straight into an even VGPR pair.
// first: overwrite H (acc starts at 0).  last: fuse bias + ELU into epilogue.
__global__ __launch_bounds__(256) void cheb_gemm(const float* __restrict__ A,
                                                 const float* __restrict__ Wk,
                                                 const float* __restrict__ bias,
                                                 float* __restrict__ H,
                                                 int rows, int first, int last) {
  __shared__ float sW[CIN * COUT];   // [CIN/2][COUT][2] interleaved
  for (int i = threadIdx.x; i < CIN * COUT; i += blockDim.x) {
    int kr  = i / COUT;
    int col = i - kr * COUT;
    sW[(((kr >> 1) * COUT) + col) * 2 + (kr & 1)] = Wk[i];
  }
  __syncthreads();

  int lane = threadIdx.x & 31;
  int wave = threadIdx.x >> 5;
  int tile = blockIdx.x * (blockDim.x >> 5) + wave;
  int row0 = tile * 16;
  if (row0 >= rows) return;

  const int mr    = lane & 15;   // M for A, N for B/C/D
  const int hi    = lane >> 4;   // lane-half select
  const int khalf = hi * 2;      // K offset 0 or 2 within a K4 step

  v8f acc0 = {}, acc1 = {}, acc2 = {}, acc3 = {};
  if (!first) {
#pragma unroll
    for (int r = 0; r < 8; ++r) {
      size_t row = (size_t)(row0 + r + hi * 8) * COUT;
      acc0[r] = H[row +  0 + mr];
      acc1[r] = H[row + 16 + mr];
      acc2[r] = H[row + 32 + mr];
      acc3[r] = H[row + 48 + mr];
    }
  }

  const float* arow = A + (size_t)(row0 + mr) * CIN;
#pragma unroll
  for (int kk = 0; kk < CIN / 4; ++kk) {
    // A 16x4 tile: a.x -> K = kk*4+khalf, a.y -> K = kk*4+khalf+1 (ISA A-layout)
    v2f a = *(const v2f*)(arow + kk * 4 + khalf);
    // B 4x16 tiles: pair index (kr/2) = kk*2 + hi; one b64 LDS load per fragment
    const float* wp = sW + (size_t)(kk * 2 + hi) * COUT * 2;
    v2f b0 = *(const v2f*)(wp + ( 0 + mr) * 2);
    v2f b1 = *(const v2f*)(wp + (16 + mr) * 2);
    v2f b2 = *(const v2f*)(wp + (32 + mr) * 2);
    v2f b3 = *(const v2f*)(wp + (48 + mr) * 2);
    acc0 = __builtin_amdgcn_wmma_f32_16x16x4_f32(false, a, false, b0, (short)0, acc0, false, false);
    acc1 = __builtin_amdgcn_wmma_f32_16x16x4_f32(false, a, false, b1, (short)0, acc1, false, false);
    acc2 = __builtin_amdgcn_wmma_f32_16x16x4_f32(false, a, false, b2, (short)0, acc2, false, false);
    acc3 = __builtin_amdgcn_wmma_f32_16x16x4_f32(false, a, false, b3, (short)0, acc3, false, false);
  }

#pragma unroll
  for (int r = 0; r < 8; ++r) {
    size_t row = (size_t)(row0 + r + hi * 8) * COUT;
    float v0 = acc0[r], v1 = acc1[r], v2 = acc2[r], v3 = acc3[r];
    if (last) {
      v0 += bias[ 0 + mr]; v1 += bias[16 + mr];
      v2 += bias[32 + mr]; v3 += bias[48 + mr];
      v0 = v0 > 0.f ? v0 : expm1f(v0);
      v1 = v1 > 0.f ? v1 : expm1f(v1);
      v2 = v2 > 0.f ? v2 : expm1f(v2);
      v3 = v3 > 0.f ? v3 : expm1f(v3);
    }
    H[row +  0 + mr] = v0;
    H[row + 16 + mr] = v1;
    H[row + 32 + mr] = v2;
    H[row + 48 + mr] = v3;
  }
}

// ---------------- pool scatter: out[b,row[j],:] += val[j]*h[b,col[j],:] ----------------
__global__ void pool_scatter(const float* __restrict__ h, const int* __restrict__ prow,
                             const int* __restrict__ pcol, const float* __restrict__ pval,
                             float* __restrict__ out, int nnz, int nN, int mPool) {
  long long t = (long long)blockIdx.x * blockDim.x + threadIdx.x;
  long long total = (long long)nnz * BATCH * (COUT / 4);
  if (t >= total) return;
  int c = (int)(t & 15) * 4;
  long long jb = t >> 4;
  int j = (int)(jb % nnz);
  int b = (int)(jb / nnz);
  float v = pval[j];
  const float4 hv = *(const float4*)(h + ((size_t)b * nN + pcol[j]) * COUT + c);
  float* d = out + ((size_t)b * mPool + prow[j]) * COUT + c;
  atomicAdd(d + 0, hv.x * v);
  atomicAdd(d + 1, hv.y * v);
  atomicAdd(d + 2, hv.z * v);
  atomicAdd(d + 3, hv.w * v);
}

extern "C" void kernel_launch(void* const* d_in, const int* in_sizes, int n_in,
                              void* d_out, int out_size, void* d_ws, size_t ws_size,
                              hipStream_t stream) {
  const float* x    = (const float*)d_in[0];
  const int*   eidx = (const int*)d_in[1];
  const float* W    = (const float*)d_in[2];
  const float* bias = (const float*)d_in[3];
  const int*   prow = (const int*)d_in[4];
  const int*   pcol = (const int*)d_in[5];
  const float* pval = (const float*)d_in[6];
  float* out = (float*)d_out;

  const int nE  = in_sizes[1] / 2;
  const int nN  = in_sizes[0] / (BATCH * CIN);
  const int nnz = in_sizes[4];
  const int mP  = out_size / (BATCH * COUT);
  const int* erow = eidx;
  const int* ecol = eidx + nE;

  const size_t rows = (size_t)BATCH * nN;   // 800000
  const size_t tsz  = rows * CIN;           // floats per T buffer

  char* ws = (char*)d_ws;
  size_t off = 0;
  auto alloc = [&](size_t nfloats) {
    float* p = (float*)(ws + off);
    off += ((nfloats * 4 + 255) & ~(size_t)255);
    return p;
  };
  float* deg  = alloc(nN);
  float* norm = alloc(nE);
  float* T[3]; T[0] = alloc(tsz); T[1] = alloc(tsz); T[2] = alloc(tsz);
  float* H    = alloc(rows * COUT);
  (void)ws_size; (void)n_in;

  const int BT = 256;

  // ---- degree -> dinv -> edge norms ----
  hipMemsetAsync(deg, 0, (size_t)nN * 4, stream);
  deg_count  <<<(nE + BT - 1) / BT, BT, 0, stream>>>(erow, deg, nE);
  deg_to_dinv<<<(nN + BT - 1) / BT, BT, 0, stream>>>(deg, nN);
  edge_norm  <<<(nE + BT - 1) / BT, BT, 0, stream>>>(erow, ecol, deg, norm, nE);

  const int tiles   = (int)(rows / 16);
  const int gblocks = (tiles + 7) / 8;

  // k = 0 : H = x @ W0
  cheb_gemm<<<gblocks, 256, 0, stream>>>(x, W, bias, H, (int)rows, 1, 0);

  // k = 1 : T1 = spmm(x); H += T1 @ W1
  long long sthreads = (long long)nE * BATCH * 8;
  int sblocks = (int)((sthreads + BT - 1) / BT);
  hipMemsetAsync(T[0], 0, tsz * 4, stream);
  spmm_scatter<<<sblocks, BT, 0, stream>>>(x, T[0], erow, ecol, norm, 1.0f, nE, nN);
  cheb_gemm<<<gblocks, 256, 0, stream>>>(T[0], W + CIN * COUT, bias, H, (int)rows, 0, 0);

  // k = 2..5 : Tk = 2*spmm(T_{k-1}) - T_{k-2}; H += Tk @ Wk (last fuses bias+ELU)
  const float* prv = x;
  float* cur = T[0];
  int nb = 1;
  long long n4 = (long long)(tsz / 4);
  int ablocks = (int)((n4 + BT - 1) / BT);
  for (int k = 2; k < KCHEB; ++k) {
    float* nxt = T[nb];
    nb = (nb + 1) % 3;
    hipMemsetAsync(nxt, 0, tsz * 4, stream);
    spmm_scatter<<<sblocks, BT, 0, stream>>>(cur, nxt, erow, ecol, norm, 2.0f, nE, nN);
    axpy_sub<<<ablocks, BT, 0, stream>>>(nxt, prv, n4);
    cheb_gemm<<<gblocks, 256, 0, stream>>>(nxt, W + k * CIN * COUT, bias, H, (int)rows, 0,
                                           (k == KCHEB - 1) ? 1 : 0);
    prv = cur; cur = nxt;
  }

  // ---- pool ----
  hipMemsetAsync(out, 0, (size_t)out_size * 4, stream);
  long long pthreads = (long long)nnz * BATCH * (COUT / 4);
  int pblocks = (int)((pthreads + BT - 1) / BT);
  pool_scatter<<<pblocks, BT, 0, stream>>>(H, prow, pcol, pval, out, nnz, nN, mP);
}